// BatchTopKSAE_64544768524323
// MI455X (gfx1250) — compile-verified
//
#include <hip/hip_runtime.h>
#include <hip/hip_bf16.h>
#include <stdint.h>

#define ACT   2048
#define DICT  16384
#define BATCH 2048

typedef float v2f __attribute__((ext_vector_type(2)));
typedef float v8f __attribute__((ext_vector_type(8)));

// -------------------- kernel 1: xm = x - b_dec --------------------
__global__ void prep_xm(const float* __restrict__ x, const float* __restrict__ bdec,
                        float* __restrict__ xm, int n) {
    int i = blockIdx.x * blockDim.x + threadIdx.x;
    if (i < n) xm[i] = x[i] - bdec[i & (ACT - 1)];
}

__global__ void zero_u32(unsigned* __restrict__ p, int n) {
    int i = blockIdx.x * blockDim.x + threadIdx.x;
    if (i < n) p[i] = 0u;
}

// -------------------- kernel 2: encoder GEMM (f32 WMMA, async double-buffered) ----------
// post[m][n] = relu( sum_k xm[m][k] * Wenc[n][k] + benc[n] )
// 256 threads = 8 waves; block tile 128(M) x 128(N); K chunk 16 via LDS double buffer.
// Wave grid 2x4 -> each wave owns 64(M) x 32(N) = 4x2 WMMA 16x16 tiles.
// Tiles fetched with GLOBAL_LOAD_ASYNC_TO_LDS_B128 (ASYNCcnt), overlapping WMMA compute.
#define BM  128
#define BN  128
#define BKT 16
#define LDA 20   // padded LDS row stride (floats): 80B rows => 16B aligned, conflict-free

#define ASYNC_CP_B128(ldsoff, base, goff)                              \
    asm volatile("global_load_async_to_lds_b128 %0, %1, %2"            \
                 :: "v"(ldsoff), "v"(goff), "s"(base) : "memory")

__global__ __launch_bounds__(256)
void sae_encode_wmma(const float* __restrict__ xm, const float* __restrict__ Wenc,
                     const float* __restrict__ benc, float* __restrict__ post) {
    __shared__ float As[2][BM * LDA];
    __shared__ float Bs[2][BN * LDA];
    const int bn0  = blockIdx.x * BN;
    const int bm0  = blockIdx.y * BM;
    const int tid  = threadIdx.x;
    const int wave = tid >> 5;
    const int lane = tid & 31;
    const int wm   = (wave >> 2) * 64;   // wave M offset (0,64)
    const int wn   = (wave & 3) * 32;    // wave N offset (0,32,64,96)
    const int lr   = lane & 15;
    const int lh   = lane >> 4;          // half-wave selector

    // per-thread cooperative-load coordinates (two b128 per matrix per chunk)
    const int r0 = tid >> 2;                  // 0..63
    const int c0 = (tid & 3) << 2;            // 0,4,8,12
    const int r1 = (tid + 256) >> 2;          // 64..127
    const int c1 = c0;
    const unsigned ldsOff0 = (unsigned)(r0 * LDA + c0) * 4u;   // LDS byte offsets in tile
    const unsigned ldsOff1 = (unsigned)(r1 * LDA + c1) * 4u;
    const unsigned ldsA[2] = { (unsigned)(size_t)&As[0][0], (unsigned)(size_t)&As[1][0] };
    const unsigned ldsB[2] = { (unsigned)(size_t)&Bs[0][0], (unsigned)(size_t)&Bs[1][0] };
    const unsigned gA0 = (unsigned)((bm0 + r0) * ACT + c0) * 4u;  // global byte offsets (kt=0)
    const unsigned gA1 = (unsigned)((bm0 + r1) * ACT + c1) * 4u;
    const unsigned gB0 = (unsigned)((bn0 + r0) * ACT + c0) * 4u;
    const unsigned gB1 = (unsigned)((bn0 + r1) * ACT + c1) * 4u;

    v8f zero = {};
    v8f acc[4][2];
#pragma unroll
    for (int a = 0; a < 4; ++a)
#pragma unroll
        for (int b = 0; b < 2; ++b) acc[a][b] = zero;

    constexpr int NC = ACT / BKT;   // 128 K-chunks

    // prologue: fetch chunk 0 into buffer 0
    {
        ASYNC_CP_B128(ldsA[0] + ldsOff0, xm,   gA0);
        ASYNC_CP_B128(ldsA[0] + ldsOff1, xm,   gA1);
        ASYNC_CP_B128(ldsB[0] + ldsOff0, Wenc, gB0);
        ASYNC_CP_B128(ldsB[0] + ldsOff1, Wenc, gB1);
    }

    for (int c = 0; c < NC; ++c) {
        const int cur = c & 1;
        if (c + 1 < NC) {
            const int nxt = cur ^ 1;
            const unsigned kb = (unsigned)(c + 1) * (BKT * 4u);
            ASYNC_CP_B128(ldsA[nxt] + ldsOff0, xm,   gA0 + kb);
            ASYNC_CP_B128(ldsA[nxt] + ldsOff1, xm,   gA1 + kb);
            ASYNC_CP_B128(ldsB[nxt] + ldsOff0, Wenc, gB0 + kb);
            ASYNC_CP_B128(ldsB[nxt] + ldsOff1, Wenc, gB1 + kb);
            asm volatile("s_wait_asynccnt 4" ::: "memory");   // current chunk landed
        } else {
            asm volatile("s_wait_asynccnt 0" ::: "memory");
        }
        __syncthreads();   // all waves' async writes visible

#pragma unroll
        for (int ks = 0; ks < BKT; ks += 4) {
            // f32 16x16x4 A-frag: lanes 0-15 hold K=ks,ks+1; lanes 16-31 hold K=ks+2,ks+3
            const int kh = ks + (lh << 1);
            v2f bf[2];
#pragma unroll
            for (int tn = 0; tn < 2; ++tn)
                bf[tn] = *(const v2f*)(&Bs[cur][(wn + tn * 16 + lr) * LDA + kh]);
#pragma unroll
            for (int tm = 0; tm < 4; ++tm) {
                v2f af = *(const v2f*)(&As[cur][(wm + tm * 16 + lr) * LDA + kh]);
#pragma unroll
                for (int tn = 0; tn < 2; ++tn)
                    acc[tm][tn] = __builtin_amdgcn_wmma_f32_16x16x4_f32(
                        false, af, false, bf[tn], (short)0, acc[tm][tn], false, false);
            }
        }
        __syncthreads();   // compute done; buffer may be overwritten next iteration
    }

    // epilogue: + b_enc, ReLU, store. C layout: VGPR v -> row (v + 8*lh), lane lr -> col
#pragma unroll
    for (int tm = 0; tm < 4; ++tm) {
#pragma unroll
        for (int tn = 0; tn < 2; ++tn) {
            int n = bn0 + wn + tn * 16 + lr;
            float be = benc[n];
#pragma unroll
            for (int v = 0; v < 8; ++v) {
                int m = bm0 + wm + tm * 16 + v + 8 * lh;
                post[(size_t)m * DICT + n] = fmaxf(acc[tm][tn][v] + be, 0.0f);
            }
        }
    }
}

// -------------------- kernels 3-6: deterministic global top-K threshold --------------------
// Positive f32 bit patterns are monotonic. Coarse bin = bits>>17 (14b), fine = (bits>>4)&8191.
__global__ __launch_bounds__(256)
void hist_coarse(const float* __restrict__ post, unsigned* __restrict__ gh) {
    __shared__ unsigned lh[16384];
    for (int i = threadIdx.x; i < 16384; i += 256) lh[i] = 0u;
    __syncthreads();
    const size_t n = (size_t)BATCH * DICT;
    const size_t stride = (size_t)gridDim.x * 256u;
    for (size_t i = (size_t)blockIdx.x * 256u + threadIdx.x; i < n; i += stride) {
        unsigned u = __float_as_uint(post[i]);
        if ((int)u > 0) atomicAdd(&lh[u >> 17], 1u);
    }
    __syncthreads();
    for (int i = threadIdx.x; i < 16384; i += 256)
        if (lh[i]) atomicAdd(&gh[i], lh[i]);
}

__global__ void scan_coarse(const unsigned* __restrict__ gh, const int* __restrict__ kptr,
                            unsigned* __restrict__ ctrl) {
    if (threadIdx.x == 0) {
        const unsigned K = (unsigned)kptr[0] * (unsigned)BATCH;
        unsigned acc = 0, binc = 0;
        int b = 16383;
        for (; b >= 0; --b) { binc = gh[b]; acc += binc; if (acc >= K) break; }
        if (b < 0) b = 0;
        ctrl[0] = (unsigned)b;        // coarse threshold bin
        ctrl[1] = acc - binc;         // count strictly above coarse bin (< K)
        ctrl[3] = K;
    }
}

__global__ __launch_bounds__(256)
void hist_fine(const float* __restrict__ post, const unsigned* __restrict__ ctrl,
               unsigned* __restrict__ gh2) {
    __shared__ unsigned lh[8192];
    for (int i = threadIdx.x; i < 8192; i += 256) lh[i] = 0u;
    __syncthreads();
    const unsigned cb = ctrl[0];
    const size_t n = (size_t)BATCH * DICT;
    const size_t stride = (size_t)gridDim.x * 256u;
    for (size_t i = (size_t)blockIdx.x * 256u + threadIdx.x; i < n; i += stride) {
        unsigned u = __float_as_uint(post[i]);
        if ((int)u > 0 && (u >> 17) == cb) atomicAdd(&lh[(u >> 4) & 8191u], 1u);
    }
    __syncthreads();
    for (int i = threadIdx.x; i < 8192; i += 256)
        if (lh[i]) atomicAdd(&gh2[i], lh[i]);
}

__global__ void scan_fine(const unsigned* __restrict__ gh2, unsigned* __restrict__ ctrl) {
    if (threadIdx.x == 0) {
        const unsigned K = ctrl[3];
        const unsigned greater = ctrl[1];
        const unsigned need = (K > greater) ? (K - greater) : 0u;
        unsigned thr;
        if (need == 0u) {
            thr = (ctrl[0] + 1u) << 17;           // keep only strictly-above bins
        } else {
            unsigned acc = 0; int b = 8191;
            for (; b >= 0; --b) { acc += gh2[b]; if (acc >= need) break; }
            if (b < 0) b = 0;
            thr = (ctrl[0] << 17) | ((unsigned)b << 4);
        }
        ctrl[2] = thr;                            // keep iff bits >= thr (and > 0)
    }
}

// -------------------- kernel 7: sparse decode --------------------
// out[b][:] = b_dec + sum over kept f of post[b][f] * Wenc[f][:]   (Wenc row f == Wdec col f)
#define MAXNNZ 4096

__global__ __launch_bounds__(256)
void sae_decode(const float* __restrict__ post, const float* __restrict__ Wenc,
                const float* __restrict__ bdec, const unsigned* __restrict__ ctrl,
                float* __restrict__ out) {
    __shared__ unsigned soff[257];
    __shared__ unsigned sidx[MAXNNZ];
    __shared__ float    sval[MAXNNZ];
    const int row = blockIdx.x;
    const int tid = threadIdx.x;
    const unsigned thr = ctrl[2];
    const float* prow = post + (size_t)row * DICT;

    // phase 1: count survivors per thread
    unsigned cnt = 0;
    for (int f = tid; f < DICT; f += 256) {
        unsigned u = __float_as_uint(prow[f]);
        if ((int)u > 0 && u >= thr) ++cnt;
    }
    soff[tid] = cnt;
    __syncthreads();
    if (tid == 0) {                       // serial exclusive scan: deterministic order
        unsigned tot = 0;
        for (int i = 0; i < 256; ++i) { unsigned c = soff[i]; soff[i] = tot; tot += c; }
        soff[256] = tot;
    }
    __syncthreads();

    // phase 2: ordered compaction into LDS
    unsigned off = soff[tid];
    for (int f = tid; f < DICT; f += 256) {
        float v = prow[f];
        unsigned u = __float_as_uint(v);
        if ((int)u > 0 && u >= thr) {
            if (off < MAXNNZ) { sidx[off] = (unsigned)f; sval[off] = v; }
            ++off;
        }
    }
    __syncthreads();
    unsigned nnz = soff[256]; if (nnz > MAXNNZ) nnz = MAXNNZ;

    // phase 3: accumulate; thread owns cols tid + 256*j (coalesced Wenc rows)
    float accv[8];
#pragma unroll
    for (int j = 0; j < 8; ++j) accv[j] = bdec[tid + j * 256];
    for (unsigned i = 0; i < nnz; ++i) {
        const unsigned f = sidx[i];
        const float v = sval[i];
        const float* w = Wenc + (size_t)f * ACT;
#pragma unroll
        for (int j = 0; j < 8; ++j) accv[j] = fmaf(v, w[tid + j * 256], accv[j]);
    }
#pragma unroll
    for (int j = 0; j < 8; ++j) out[(size_t)row * ACT + tid + j * 256] = accv[j];
}

// -------------------- launch --------------------
extern "C" void kernel_launch(void* const* d_in, const int* in_sizes, int n_in,
                              void* d_out, int out_size, void* d_ws, size_t ws_size,
                              hipStream_t stream) {
    const float* x    = (const float*)d_in[0];   // (BATCH, ACT)
    const float* Wenc = (const float*)d_in[1];   // (DICT, ACT)
    const float* benc = (const float*)d_in[2];   // (DICT,)
    /* d_in[3] = W_dec is (ACT, DICT) = Wenc^T; we use Wenc rows for contiguous decode */
    const float* bdec = (const float*)d_in[4];   // (ACT,)
    const int*   kptr = (const int*)d_in[5];     // scalar k

    float*    post = (float*)d_ws;                                   // BATCH*DICT f32
    float*    xm   = post + (size_t)BATCH * DICT;                    // BATCH*ACT f32
    unsigned* h1   = (unsigned*)(xm + (size_t)BATCH * ACT);          // 16384 u32
    unsigned* h2   = h1 + 16384;                                     // 8192 u32
    unsigned* ctrl = h2 + 8192;                                      // 8 u32

    prep_xm<<<(BATCH * ACT + 255) / 256, 256, 0, stream>>>(x, bdec, xm, BATCH * ACT);
    zero_u32<<<(16384 + 8192 + 8 + 255) / 256, 256, 0, stream>>>(h1, 16384 + 8192 + 8);

    dim3 g(DICT / BN, BATCH / BM);
    sae_encode_wmma<<<g, 256, 0, stream>>>(xm, Wenc, benc, post);

    hist_coarse<<<2048, 256, 0, stream>>>(post, h1);
    scan_coarse<<<1, 32, 0, stream>>>(h1, kptr, ctrl);
    hist_fine<<<2048, 256, 0, stream>>>(post, ctrl, h2);
    scan_fine<<<1, 32, 0, stream>>>(h2, ctrl);

    sae_decode<<<BATCH, 256, 0, stream>>>(post, Wenc, bdec, ctrl, (float*)d_out);
}